// CompGCNLayer_39762807226645
// MI455X (gfx1250) — compile-verified
//
#include <hip/hip_runtime.h>
#include <hip/hip_bf16.h>

#define DIM 128
#define BN_EPS 1e-5f

typedef __attribute__((ext_vector_type(2))) float v2f;
typedef __attribute__((ext_vector_type(8))) float v8f;

__device__ __forceinline__ v8f wmma_f32_k4(v2f a, v2f b, v8f c) {
    // D = A(16x4) * B(4x16) + C ; emits v_wmma_f32_16x16x4_f32
    return __builtin_amdgcn_wmma_f32_16x16x4_f32(
        /*neg_a=*/false, a, /*neg_b=*/false, b,
        /*c_mod=*/(short)0, c, /*reuse_a=*/false, /*reuse_b=*/false);
}

__device__ __forceinline__ void atomic_add_f32(float* p, float v) {
    __hip_atomic_fetch_add(p, v, __ATOMIC_RELAXED, __HIP_MEMORY_SCOPE_AGENT);
}
__device__ __forceinline__ int atomic_add_i32(int* p, int v) {
    return __hip_atomic_fetch_add(p, v, __ATOMIC_RELAXED, __HIP_MEMORY_SCOPE_AGENT);
}

// ---------------------------------------------------------------- zero scratch
__global__ void zero_kernel(int* __restrict__ p, size_t n) {
    size_t i = (size_t)blockIdx.x * blockDim.x + threadIdx.x;
    size_t stride = (size_t)gridDim.x * blockDim.x;
    for (; i < n; i += stride) p[i] = 0;
}

// ---------------------------------------------------------------- degrees
__global__ void degree_kernel(const int* __restrict__ ei, int nE,
                              int* __restrict__ deg0, int* __restrict__ deg1) {
    int e = blockIdx.x * blockDim.x + threadIdx.x;
    if (e < nE) {
        atomic_add_i32(&deg0[ei[e]], 1);       // counts over edge_index[0] (src)
        atomic_add_i32(&deg1[ei[nE + e]], 1);  // counts over edge_index[1] (tgt)
    }
}

// ------------------------------------------------ exclusive prefix scan (CSR)
// single workgroup: chunked sums -> LDS Hillis-Steele -> per-chunk write-out
__global__ void __launch_bounds__(1024)
scan_kernel(const int* __restrict__ deg, int n, int* __restrict__ rowStart) {
    __shared__ int part[1024];
    const int t = threadIdx.x;
    const int chunk = (n + 1023) >> 10;
    const int base = t * chunk;
    int s = 0;
    for (int i = 0; i < chunk; ++i) {
        int idx = base + i;
        if (idx < n) s += deg[idx];
    }
    part[t] = s;
    __syncthreads();
    for (int off = 1; off < 1024; off <<= 1) {
        int v = (t >= off) ? part[t - off] : 0;
        __syncthreads();
        part[t] += v;
        __syncthreads();
    }
    int excl = (t == 0) ? 0 : part[t - 1];
    for (int i = 0; i < chunk; ++i) {
        int idx = base + i;
        if (idx < n) { rowStart[idx] = excl; excl += deg[idx]; }
    }
    if (t == 1023) rowStart[n] = part[1023];
}

// ---------------------------------------------------------------- CSR fill
__global__ void fill_kernel(const int* __restrict__ ei, int nE,
                            const int* __restrict__ rsF, const int* __restrict__ rsB,
                            int* __restrict__ cntF, int* __restrict__ cntB,
                            int* __restrict__ listF, int* __restrict__ listB) {
    int e = blockIdx.x * blockDim.x + threadIdx.x;
    if (e >= nE) return;
    int s = ei[e];
    int t = ei[nE + e];
    listF[rsF[t] + atomic_add_i32(&cntF[t], 1)] = e;  // fwd keyed by tgt
    listB[rsB[s] + atomic_add_i32(&cntB[s], 1)] = e;  // bwd keyed by src
}

// -------------------------------------------- gather-accumulate (NO atomics)
// one wave32 per node; lane holds 4 dims; writes the h row exactly once
__global__ void __launch_bounds__(256)
gather_accum(const float* __restrict__ xe, const float* __restrict__ xr,
             const int* __restrict__ ei, const int* __restrict__ et,
             const int* __restrict__ deg0, const int* __restrict__ deg1,
             const int* __restrict__ rowStart, const int* __restrict__ list,
             float* __restrict__ h, int nE, int nEnt, int relOff, int otherRow0) {
    int node = blockIdx.x * 8 + (threadIdx.x >> 5);
    if (node >= nEnt) return;
    const int lane = threadIdx.x & 31;
    const int beg = rowStart[node];
    const int end = rowStart[node + 1];
    float4 acc = {0.f, 0.f, 0.f, 0.f};
    for (int j = beg; j < end; ++j) {
        if (j + 8 < end) __builtin_prefetch(&list[j + 8], 0, 0);  // global_prefetch_b8
        int e   = list[j];
        int src = ei[e];
        int tgt = ei[nE + e];
        float w = rsqrtf((float)deg0[src] * (float)deg1[tgt]);  // symmetric weight
        int other = otherRow0 ? src : tgt;
        int rel   = et[e] * 2 + relOff;
        float4 a = ((const float4*)(xe + (size_t)other * DIM))[lane];
        float4 r = ((const float4*)(xr + (size_t)rel * DIM))[lane];
        acc.x += w * (a.x - r.x);
        acc.y += w * (a.y - r.y);
        acc.z += w * (a.z - r.z);
        acc.w += w * (a.w - r.w);
    }
    ((float4*)(h + (size_t)node * DIM))[lane] = acc;   // single clean b128 store
}

// ------------------------------------------- fused 3-way WMMA GEMM (pre-BN out)
// out = ((x_e - self_loop)@Wl + hf@Wf + hb@Wb) / 3 + bias
#define LDSP 132  // 16B-aligned pad: bank = 4*m + k -> conflict-free
__global__ void __launch_bounds__(256)
gemm_nodes(const float* __restrict__ xe, const float* __restrict__ hf,
           const float* __restrict__ hb, const float* __restrict__ wl,
           const float* __restrict__ wf, const float* __restrict__ wb,
           const float* __restrict__ self_loop, const float* __restrict__ bias,
           float* __restrict__ out, int nEnt) {
    __shared__ float sA0[16 * LDSP];
    __shared__ float sA1[16 * LDSP];
    __shared__ float sA2[16 * LDSP];

    const int tid  = threadIdx.x;
    const int row0 = blockIdx.x * 16;
    const int lr   = tid >> 4;          // 0..15 row of tile
    const int lc   = (tid & 15) * 8;    // 8 consecutive cols per thread
    const int grow = row0 + lr;
    const bool rok = grow < nEnt;

    float4 z = {0.f, 0.f, 0.f, 0.f};
    const float4 s0 = ((const float4*)(self_loop + lc))[0];
    const float4 s1 = ((const float4*)(self_loop + lc))[1];

    float4 v0 = z, v1 = z;
    if (rok) {
        const float* g = xe + (size_t)grow * DIM + lc;
        v0 = ((const float4*)g)[0]; v1 = ((const float4*)g)[1];
    }
    v0.x -= s0.x; v0.y -= s0.y; v0.z -= s0.z; v0.w -= s0.w;
    v1.x -= s1.x; v1.y -= s1.y; v1.z -= s1.z; v1.w -= s1.w;
    ((float4*)&sA0[lr * LDSP + lc])[0] = v0;
    ((float4*)&sA0[lr * LDSP + lc])[1] = v1;

    v0 = z; v1 = z;
    if (rok) {
        const float* g = hf + (size_t)grow * DIM + lc;
        v0 = ((const float4*)g)[0]; v1 = ((const float4*)g)[1];
    }
    ((float4*)&sA1[lr * LDSP + lc])[0] = v0;
    ((float4*)&sA1[lr * LDSP + lc])[1] = v1;

    v0 = z; v1 = z;
    if (rok) {
        const float* g = hb + (size_t)grow * DIM + lc;
        v0 = ((const float4*)g)[0]; v1 = ((const float4*)g)[1];
    }
    ((float4*)&sA2[lr * LDSP + lc])[0] = v0;
    ((float4*)&sA2[lr * LDSP + lc])[1] = v1;

    __syncthreads();

    const int wave = tid >> 5;
    const int lane = tid & 31;
    const int m    = lane & 15;       // A row / D col-lane
    const int half = lane >> 4;       // K-group selector
    const int n    = wave * 16 + m;   // output column

    const float* A0 = &sA0[m * LDSP];
    const float* A1 = &sA1[m * LDSP];
    const float* A2 = &sA2[m * LDSP];

    v8f c = {};
    #pragma unroll
    for (int k0 = 0; k0 < DIM; k0 += 4) {
        const int kk = k0 + 2 * half;
        v2f a, b;
        a.x = A0[kk]; a.y = A0[kk + 1];
        b.x = wl[(size_t)kk * DIM + n]; b.y = wl[(size_t)(kk + 1) * DIM + n];
        c = wmma_f32_k4(a, b, c);
        a.x = A1[kk]; a.y = A1[kk + 1];
        b.x = wf[(size_t)kk * DIM + n]; b.y = wf[(size_t)(kk + 1) * DIM + n];
        c = wmma_f32_k4(a, b, c);
        a.x = A2[kk]; a.y = A2[kk + 1];
        b.x = wb[(size_t)kk * DIM + n]; b.y = wb[(size_t)(kk + 1) * DIM + n];
        c = wmma_f32_k4(a, b, c);
    }

    const float bn = bias[n];
    #pragma unroll
    for (int r = 0; r < 8; ++r) {
        int row = row0 + r + 8 * half;    // D layout: VGPR r -> M = r + 8*half
        if (row < nEnt)
            out[(size_t)row * DIM + n] = c[r] * (1.0f / 3.0f) + bn;
    }
}

// ---------------------------------------------------------------- BN stats
__global__ void __launch_bounds__(256)
bn_stats(const float* __restrict__ out, int nEnt,
         float* __restrict__ colSum, float* __restrict__ colSq) {
    __shared__ float ss[256];
    __shared__ float sq[256];
    const int col = threadIdx.x & 127;
    const int rl  = threadIdx.x >> 7;
    float s = 0.f, s2 = 0.f;
    for (int row = blockIdx.x * 2 + rl; row < nEnt; row += gridDim.x * 2) {
        float v = out[(size_t)row * DIM + col];
        s += v; s2 += v * v;
    }
    ss[threadIdx.x] = s; sq[threadIdx.x] = s2;
    __syncthreads();
    if (threadIdx.x < 128) {
        atomic_add_f32(&colSum[col], ss[threadIdx.x] + ss[threadIdx.x + 128]);
        atomic_add_f32(&colSq[col],  sq[threadIdx.x] + sq[threadIdx.x + 128]);
    }
}

// ---------------------------------------------------------------- BN apply
__global__ void bn_apply(float* __restrict__ out, size_t n, float invN,
                         const float* __restrict__ colSum, const float* __restrict__ colSq,
                         const float* __restrict__ gamma, const float* __restrict__ beta) {
    size_t i = (size_t)blockIdx.x * blockDim.x + threadIdx.x;
    size_t stride = (size_t)gridDim.x * blockDim.x;
    for (; i < n; i += stride) {
        int col = (int)(i & (DIM - 1));
        float mean = colSum[col] * invN;
        float var  = colSq[col] * invN - mean * mean;
        float v = out[i];
        out[i] = (v - mean) * rsqrtf(var + BN_EPS) * gamma[col] + beta[col];
    }
}

// --------------------------------------------- relation GEMM: x_r @ w_rel^T
__global__ void __launch_bounds__(256)
gemm_rel(const float* __restrict__ xr, const float* __restrict__ wrel,
         float* __restrict__ outr, int nR) {
    __shared__ float sA[16 * LDSP];
    const int tid  = threadIdx.x;
    const int row0 = blockIdx.x * 16;
    const int lr   = tid >> 4;
    const int lc   = (tid & 15) * 8;

    float4 v0 = {0.f, 0.f, 0.f, 0.f}, v1 = v0;
    if (row0 + lr < nR) {
        const float* g = xr + (size_t)(row0 + lr) * DIM + lc;
        v0 = ((const float4*)g)[0]; v1 = ((const float4*)g)[1];
    }
    ((float4*)&sA[lr * LDSP + lc])[0] = v0;
    ((float4*)&sA[lr * LDSP + lc])[1] = v1;
    __syncthreads();

    const int wave = tid >> 5;
    const int lane = tid & 31;
    const int m    = lane & 15;
    const int half = lane >> 4;
    const int n    = wave * 16 + m;

    const float* A = &sA[m * LDSP];
    const float* B = wrel + (size_t)n * DIM;  // B[k][n] = wrel[n][k] -> contiguous in k

    v8f c = {};
    #pragma unroll
    for (int k0 = 0; k0 < DIM; k0 += 4) {
        const int kk = k0 + 2 * half;
        v2f a, b;
        a.x = A[kk]; a.y = A[kk + 1];
        float2 bl = *((const float2*)(B + kk));
        b.x = bl.x; b.y = bl.y;
        c = wmma_f32_k4(a, b, c);
    }

    #pragma unroll
    for (int r = 0; r < 8; ++r) {
        int row = row0 + r + 8 * half;
        if (row < nR) outr[(size_t)row * DIM + n] = c[r];
    }
}

// ================================================================= launcher
extern "C" void kernel_launch(void* const* d_in, const int* in_sizes, int n_in,
                              void* d_out, int out_size, void* d_ws, size_t ws_size,
                              hipStream_t stream) {
    const float* xe    = (const float*)d_in[0];
    const float* xr    = (const float*)d_in[1];
    const float* wl    = (const float*)d_in[2];
    const float* wf    = (const float*)d_in[3];
    const float* wb    = (const float*)d_in[4];
    const float* wrel  = (const float*)d_in[5];
    const float* slf   = (const float*)d_in[6];
    const float* bias  = (const float*)d_in[7];
    const float* gamma = (const float*)d_in[8];
    const float* beta  = (const float*)d_in[9];
    const int*   eidx  = (const int*)d_in[10];
    const int*   etype = (const int*)d_in[11];

    const int nEnt = in_sizes[0] / DIM;
    const int nR   = in_sizes[1] / DIM;   // 2*N_REL rows
    const int nE   = in_sizes[11];

    float* out  = (float*)d_out;
    float* outr = out + (size_t)nEnt * DIM;

    // ---------------- workspace layout (4-byte words) ----------------
    int* wsi = (int*)d_ws;
    size_t o = 0;
    int* deg0 = wsi + o;  o += nEnt;            // zeroed region starts here
    int* deg1 = wsi + o;  o += nEnt;
    int* cntF = wsi + o;  o += nEnt;
    int* cntB = wsi + o;  o += nEnt;
    float* colSum = (float*)(wsi + o); o += DIM;
    float* colSq  = (float*)(wsi + o); o += DIM;
    const size_t zcount = o;                    // contiguous zero-init region
    int* rsF   = wsi + o; o += (size_t)nEnt + 1;
    int* rsB   = wsi + o; o += (size_t)nEnt + 1;
    int* listF = wsi + o; o += nE;
    int* listB = wsi + o; o += nE;
    o = (o + 7) & ~(size_t)7;                   // 32B-align for float4 rows
    float* hf = (float*)(wsi + o); o += (size_t)nEnt * DIM;
    float* hb = (float*)(wsi + o); o += (size_t)nEnt * DIM;

    // ---------------- pipeline ----------------
    zero_kernel<<<1024, 256, 0, stream>>>(wsi, zcount);
    degree_kernel<<<(nE + 255) / 256, 256, 0, stream>>>(eidx, nE, deg0, deg1);
    scan_kernel<<<1, 1024, 0, stream>>>(deg1, nEnt, rsF);   // fwd CSR keyed by tgt
    scan_kernel<<<1, 1024, 0, stream>>>(deg0, nEnt, rsB);   // bwd CSR keyed by src
    fill_kernel<<<(nE + 255) / 256, 256, 0, stream>>>(eidx, nE, rsF, rsB,
                                                      cntF, cntB, listF, listB);
    const int gblocks = (nEnt + 7) / 8;
    gather_accum<<<gblocks, 256, 0, stream>>>(xe, xr, eidx, etype, deg0, deg1,
                                              rsF, listF, hf, nE, nEnt,
                                              /*relOff=*/0, /*otherRow0=*/1);
    gather_accum<<<gblocks, 256, 0, stream>>>(xe, xr, eidx, etype, deg0, deg1,
                                              rsB, listB, hb, nE, nEnt,
                                              /*relOff=*/1, /*otherRow0=*/0);
    gemm_nodes<<<(nEnt + 15) / 16, 256, 0, stream>>>(xe, hf, hb, wl, wf, wb,
                                                     slf, bias, out, nEnt);
    bn_stats<<<200, 256, 0, stream>>>(out, nEnt, colSum, colSq);
    bn_apply<<<2048, 256, 0, stream>>>(out, (size_t)nEnt * DIM, 1.0f / (float)nEnt,
                                       colSum, colSq, gamma, beta);
    gemm_rel<<<(nR + 15) / 16, 256, 0, stream>>>(xr, wrel, outr, nR);
}